// PartTripletLoss_82995948028521
// MI455X (gfx1250) — compile-verified
//
#include <hip/hip_runtime.h>

typedef float v2f __attribute__((ext_vector_type(2)));
typedef float v4f __attribute__((ext_vector_type(4)));
typedef float v8f __attribute__((ext_vector_type(8)));

#define N_PARTS 64
#define M_SAMP  512
#define D_FEAT  256
#define K_POS   16
#define MARGIN  0.2f
#define A_PAD   260   // 256 + 4 pad: bank-conflict-free 16-row b64 fragment reads

// ---------------------------------------------------------------------------
// Kernel 0: per-row squared norms. One wave32 per row, shfl reduction.
// ---------------------------------------------------------------------------
__global__ void row_norms_kernel(const float* __restrict__ feat,
                                 float* __restrict__ x2) {
  int gw   = (blockIdx.x * blockDim.x + threadIdx.x) >> 5;  // global wave = row
  int lane = threadIdx.x & 31;
  if (gw >= N_PARTS * M_SAMP) return;
  const float* r = feat + (size_t)gw * D_FEAT;
  float s = 0.f;
#pragma unroll
  for (int c = 0; c < D_FEAT; c += 32) {
    float v = r[c + lane];
    s += v * v;
  }
  for (int off = 16; off > 0; off >>= 1) s += __shfl_down(s, off);
  if (lane == 0) x2[gw] = s;
}

// ---------------------------------------------------------------------------
// Kernel 1: zero the per-part accumulators (128 consecutive floats).
// ---------------------------------------------------------------------------
__global__ void zero_kernel(float* p) { p[threadIdx.x] = 0.f; }

// ---------------------------------------------------------------------------
// Kernel 2: fused dist-strip (WMMA f32 Gram) + hinge loss.
// grid = (32 strips, 64 parts), block = 256 (8 wave32).
//   - A strip (16 anchors x 256) staged once in LDS, padded stride (no bank
//     conflicts on the 16-row fragment reads).
//   - each wave computes 4 column tiles; fragment loads software-pipelined
//     one K-step ahead of the v_wmma_f32_16x16x4_f32 issues.
//   - dist strip lives in LDS; hinge loss reduced to per-part atomics.
// ---------------------------------------------------------------------------
__global__ void triplet_strip_kernel(const float* __restrict__ feat,
                                     const int*   __restrict__ label,
                                     const float* __restrict__ x2,
                                     float* __restrict__ part_sum,
                                     float* __restrict__ part_cnt) {
  __shared__ float s_a[K_POS][A_PAD];        // A strip, 16.6 KB
  __shared__ float s_dist[K_POS * M_SAMP];   // 16 x 512 strip, 32 KB
  __shared__ float s_x2col[M_SAMP];
  __shared__ float s_x2row[K_POS];
  __shared__ int   s_lab[M_SAMP];
  __shared__ int   s_labi[K_POS];
  __shared__ float s_pos[K_POS][K_POS];      // positive dists per anchor
  __shared__ int   s_pcnt[K_POS];
  __shared__ float s_tot[2];                 // block sum / count

  const int strip = blockIdx.x;              // 0..31
  const int part  = blockIdx.y;              // 0..63
  const int tid   = threadIdx.x;             // 0..255
  const int ibase = strip * K_POS;

  const float* Xp   = feat  + (size_t)part * M_SAMP * D_FEAT;
  const float* x2p  = x2    + part * M_SAMP;
  const int*   labp = label + part * M_SAMP;

  // ---- phase 1: stage x2 / labels / A strip into LDS (coalesced) ----
  for (int j = tid; j < M_SAMP; j += 256) {
    s_x2col[j] = x2p[j];
    s_lab[j]   = labp[j];
  }
  {
    const float* Arows = Xp + (size_t)ibase * D_FEAT;   // 16 x 256 contiguous
    for (int q = tid; q < K_POS * (D_FEAT / 4); q += 256) {
      int row = q >> 6;             // 64 float4 per row
      int c4  = (q & 63) << 2;
      v4f v = *(const v4f*)(Arows + row * D_FEAT + c4);
      *(v4f*)&s_a[row][c4] = v;     // row stride 1040 B: 16B aligned
    }
  }
  if (tid < K_POS) {
    s_x2row[tid] = x2p[ibase + tid];
    s_labi[tid]  = labp[ibase + tid];
    s_pcnt[tid]  = 0;
  }
  if (tid < 2) s_tot[tid] = 0.f;
  __syncthreads();

  // ---- phase 2: 16x512 Gram strip via V_WMMA_F32_16X16X4_F32 ----
  const int lane  = tid & 31;
  const int wave  = tid >> 5;
  const int rsel  = lane & 15;               // row within 16-row fragment
  const int khalf = (lane >> 4) << 1;        // lanes 16-31 hold K=2,3

  const float* brow[4];
  int jbase[4];
#pragma unroll
  for (int t = 0; t < 4; ++t) {
    jbase[t] = (wave + t * 8) * 16;
    brow[t]  = Xp + (size_t)(jbase[t] + rsel) * D_FEAT + khalf;
  }
  const float* alds = &s_a[rsel][khalf];

  v8f acc[4];
#pragma unroll
  for (int t = 0; t < 4; ++t) { v8f z = {}; acc[t] = z; }

  // software pipeline: fetch K+4 fragments before issuing K's WMMAs
  v2f a_cur = *(const v2f*)(alds);
  v2f b_cur[4];
#pragma unroll
  for (int t = 0; t < 4; ++t) b_cur[t] = *(const v2f*)(brow[t]);

  for (int k = 0; k < D_FEAT; k += 4) {
    int kn = (k + 4 < D_FEAT) ? (k + 4) : 0;
    v2f a_nxt = *(const v2f*)(alds + kn);
    v2f b_nxt[4];
#pragma unroll
    for (int t = 0; t < 4; ++t) b_nxt[t] = *(const v2f*)(brow[t] + kn);

#pragma unroll
    for (int t = 0; t < 4; ++t)
      acc[t] = __builtin_amdgcn_wmma_f32_16x16x4_f32(
          false, a_cur, false, b_cur[t], (short)0, acc[t], false, false);

    a_cur = a_nxt;
#pragma unroll
    for (int t = 0; t < 4; ++t) b_cur[t] = b_nxt[t];
  }

  // dist = sqrt(relu(x2_i + x2_j - 2*G)); C/D layout: M = v + 8*(lane>=16)
#pragma unroll
  for (int t = 0; t < 4; ++t) {
#pragma unroll
    for (int v = 0; v < 8; ++v) {
      int Mrow = v + ((lane >> 4) << 3);
      int Ncol = jbase[t] + (lane & 15);
      float d2 = s_x2row[Mrow] + s_x2col[Ncol] - 2.f * acc[t][v];
      s_dist[Mrow * M_SAMP + Ncol] = sqrtf(fmaxf(d2, 0.f));
    }
  }
  __syncthreads();

  // ---- phase 3: gather positive distances per anchor ----
  for (int j = tid; j < M_SAMP; j += 256) {
    int lj = s_lab[j];
#pragma unroll
    for (int a = 0; a < K_POS; ++a) {
      if (lj == s_labi[a]) {
        int slot = atomicAdd(&s_pcnt[a], 1);
        if (slot < K_POS) s_pos[a][slot] = s_dist[a * M_SAMP + j];
      }
    }
  }
  __syncthreads();

  // ---- phase 4: hinge loss over (pos, neg) pairs ----
  float lsum = 0.f;
  int   lcnt = 0;
  for (int a = 0; a < K_POS; ++a) {
    float pa[K_POS];
#pragma unroll
    for (int kk = 0; kk < K_POS; ++kk) pa[kk] = s_pos[a][kk];
    int la = s_labi[a];
    for (int j = tid; j < M_SAMP; j += 256) {
      if (s_lab[j] == la) continue;        // skip positives/self
      float dn = s_dist[a * M_SAMP + j];
#pragma unroll
      for (int kk = 0; kk < K_POS; ++kk) {
        float v = MARGIN + pa[kk] - dn;
        if (v > 0.f) { lsum += v; lcnt += 1; }
      }
    }
  }
  float fc = (float)lcnt;
  for (int off = 16; off > 0; off >>= 1) {
    lsum += __shfl_down(lsum, off);
    fc   += __shfl_down(fc, off);
  }
  if (lane == 0) {
    atomicAdd(&s_tot[0], lsum);
    atomicAdd(&s_tot[1], fc);
  }
  __syncthreads();
  if (tid == 0) {
    atomicAdd(&part_sum[part], s_tot[0]);
    atomicAdd(&part_cnt[part], s_tot[1]);
  }
}

// ---------------------------------------------------------------------------
// Kernel 3: per-part mean (0 when no active triplets), then mean over parts.
// ---------------------------------------------------------------------------
__global__ void finalize_kernel(const float* __restrict__ part_sum,
                                const float* __restrict__ part_cnt,
                                float* __restrict__ out) {
  __shared__ float ls[N_PARTS], lc[N_PARTS];
  int p = threadIdx.x;
  float s = part_sum[p], c = part_cnt[p];
  ls[p] = (c == 0.f) ? 0.f : s / fmaxf(c, 1.f);
  lc[p] = c;
  __syncthreads();
  if (p == 0) {
    float a = 0.f, b = 0.f;
    for (int i = 0; i < N_PARTS; ++i) { a += ls[i]; b += lc[i]; }
    out[0] = a / (float)N_PARTS;
    out[1] = b / (float)N_PARTS;
  }
}

// ---------------------------------------------------------------------------
extern "C" void kernel_launch(void* const* d_in, const int* in_sizes, int n_in,
                              void* d_out, int out_size, void* d_ws, size_t ws_size,
                              hipStream_t stream) {
  const float* feat  = (const float*)d_in[0];
  const int*   label = (const int*)d_in[1];
  // d_in[2] = num_pos (device scalar) — positives identified via labels.

  float* x2       = (float*)d_ws;                 // 64*512 floats
  float* part_sum = x2 + N_PARTS * M_SAMP;        // 64 floats
  float* part_cnt = part_sum + N_PARTS;           // 64 floats (zeroed together)
  float* out      = (float*)d_out;

  zero_kernel<<<1, 2 * N_PARTS, 0, stream>>>(part_sum);
  row_norms_kernel<<<(N_PARTS * M_SAMP * 32) / 256, 256, 0, stream>>>(feat, x2);
  triplet_strip_kernel<<<dim3(M_SAMP / K_POS, N_PARTS), 256, 0, stream>>>(
      feat, label, x2, part_sum, part_cnt);
  finalize_kernel<<<1, N_PARTS, 0, stream>>>(part_sum, part_cnt, out);
}